// Gradient_48395691491668
// MI455X (gfx1250) — compile-verified
//
#include <hip/hip_runtime.h>
#include <hip/hip_bf16.h>
#include <stdint.h>

// Pure streaming copy: d_out = concat(out, target). 128 MiB of HBM traffic,
// ~5.5us at 23.3 TB/s. Use the CDNA5 async global<->LDS DMA path
// (GLOBAL_LOAD_ASYNC_TO_LDS_B128 / GLOBAL_STORE_ASYNC_FROM_LDS_B128, ASYNCcnt)
// so data flows cache->LDS->cache with zero VGPR/VALU involvement.

#define THREADS 256
#define CHUNKS 4 // float4 chunks per thread -> 16 KiB LDS tile per block

__global__ __launch_bounds__(THREADS) void async_copy_b128(
    const float* __restrict__ src, float* __restrict__ dst, long long n /*floats*/) {
  __shared__ float4 tile[THREADS * CHUNKS];

  const int tid = threadIdx.x;
  const long long base4 = (long long)blockIdx.x * (THREADS * CHUNKS); // in float4 units

  if ((base4 + THREADS * CHUNKS) * 4 <= n) {
    // Fast path: full 16 KiB tile, 128-bit async DMA both directions.
    const float4* __restrict__ s4 = (const float4*)src;
    float4* __restrict__ d4 = (float4*)dst;
#pragma unroll
    for (int k = 0; k < CHUNKS; ++k) {
      const long long i4 = base4 + (long long)k * THREADS + tid;
      // Low 32 bits of the flat pointer to a __shared__ object == LDS byte offset.
      const uint32_t lds_off = (uint32_t)(uintptr_t)(const void*)&tile[k * THREADS + tid];
      const unsigned long long gaddr = (unsigned long long)(uintptr_t)(s4 + i4);
      asm volatile("global_load_async_to_lds_b128 %0, %1, off"
                   :
                   : "v"(lds_off), "v"(gaddr)
                   : "memory");
    }
    // Async loads complete (LDS written) when ASYNCcnt drains.
    asm volatile("s_wait_asynccnt 0" ::: "memory");
#pragma unroll
    for (int k = 0; k < CHUNKS; ++k) {
      const long long i4 = base4 + (long long)k * THREADS + tid;
      const uint32_t lds_off = (uint32_t)(uintptr_t)(const void*)&tile[k * THREADS + tid];
      const unsigned long long gaddr = (unsigned long long)(uintptr_t)(d4 + i4);
      asm volatile("global_store_async_from_lds_b128 %0, %1, off"
                   :
                   : "v"(gaddr), "v"(lds_off)
                   : "memory");
    }
    asm volatile("s_wait_asynccnt 0" ::: "memory");
  } else {
    // Tail path (never taken for the 512x512x8x4 shape, kept for generality).
#pragma unroll
    for (int k = 0; k < CHUNKS; ++k) {
      const long long i = (base4 + (long long)k * THREADS + tid) * 4;
#pragma unroll
      for (int j = 0; j < 4; ++j) {
        if (i + j < n) dst[i + j] = src[i + j];
      }
    }
  }
}

extern "C" void kernel_launch(void* const* d_in, const int* in_sizes, int n_in,
                              void* d_out, int out_size, void* d_ws, size_t ws_size,
                              hipStream_t stream) {
  (void)d_ws; (void)ws_size; (void)n_in; (void)out_size;
  // setup_inputs order: x (unused), out, target. Output = concat(out, target), fp32.
  const float* out_src = (const float*)d_in[1];
  const float* tgt_src = (const float*)d_in[2];
  float* dst = (float*)d_out;

  const long long n_out = (long long)in_sizes[1];
  const long long n_tgt = (long long)in_sizes[2];

  const long long tile_f = (long long)THREADS * CHUNKS * 4; // floats per block
  const int blocks_out = (int)((n_out + tile_f - 1) / tile_f);
  const int blocks_tgt = (int)((n_tgt + tile_f - 1) / tile_f);

  async_copy_b128<<<blocks_out, THREADS, 0, stream>>>(out_src, dst, n_out);
  async_copy_b128<<<blocks_tgt, THREADS, 0, stream>>>(tgt_src, dst + n_out, n_tgt);
}